// Net2_54494545051831
// MI455X (gfx1250) — compile-verified
//
#include <hip/hip_runtime.h>
#include <hip/hip_bf16.h>

typedef float v2f __attribute__((ext_vector_type(2)));
typedef float v8f __attribute__((ext_vector_type(8)));

#define ICC   3
#define KHH   7
#define KWW   7
#define OHH   26
#define OWW   26
#define NLOC  (OHH * OWW)          // 676
#define OCC   42
#define NCLS  10
#define NB    1024
#define KDIM  147                  // ICC*KHH*KWW
#define KPAD  148                  // padded to multiple of 4
#define CHUNK 64                   // batch rows per chunk (4 waves x 16)
#define NCHUNK (NB / CHUNK)        // 16
#define FLAT  (NLOC * OCC)         // 28392

// LDS budget (floats):
//   sW   : KPAD*OCC        = 6216   (K-pair-major: [k/2][o][2])
//   sP   : CHUNK*KPAD      = 9472   (aliased by sAct = CHUNK*OCC = 2688)
//   sLW  : NCLS*OCC        = 420
//   sB   : OCC             = 42
// total = 16150 floats = 64,600 bytes (<= 64 KB)
#define SMEM_FLOATS 16150

__global__ __launch_bounds__(128) void lc_wmma_kernel(
    const float* __restrict__ x,     // [1024,3,32,32]
    const float* __restrict__ W,     // [42,3,26,26,49]
    const float* __restrict__ bias,  // [42,26,26]
    const float* __restrict__ lw,    // [10,28392]
    float* __restrict__ partial)     // [676,1024,10] workspace
{
    __shared__ float smem[SMEM_FLOATS];
    float* sW   = smem;                      // [KPAD/2][OCC][2] K-pair-major
    float* sP   = smem + KPAD * OCC;         // [CHUNK][KPAD]
    float* sAct = sP;                        // alias: [CHUNK][OCC]
    float* sLW  = sP + CHUNK * KPAD;         // [NCLS][OCC]
    float* sB   = sLW + NCLS * OCC;          // [OCC]

    const int hw  = blockIdx.x;              // 0..675
    const int h   = hw / OWW;
    const int w   = hw % OWW;
    const int tid = threadIdx.x;             // 0..127
    const int lane = tid & 31;
    const int wv   = tid >> 5;               // wave id 0..3
    const int mrow = lane & 15;              // M (and N) index within tile
    const int khalf = lane >> 4;             // selects K pair {0,1} vs {2,3}

    // --- one-time per-location staging -------------------------------------
    // W^T, K-pair swizzled: sW[((kg/2)*OCC + o)*2 + kg%2] = W[o][c][h][w][k49],
    // kg = c*49 + k49; zero-pad kg==147. Each B fragment is then one aligned
    // 8-byte LDS load.
    for (int i = tid; i < KPAD * OCC; i += 128) {
        int kg = i / OCC, o = i % OCC;
        float v = 0.0f;
        if (kg < KDIM) {
            int c = kg / 49, k49 = kg % 49;
            v = W[(((size_t)(o * ICC + c) * OHH + h) * OWW + w) * 49 + k49];
        }
        sW[((kg >> 1) * OCC + o) * 2 + (kg & 1)] = v;
    }
    // classifier slice: sLW[cls][o] = lw[cls][o*676 + hw]
    for (int i = tid; i < NCLS * OCC; i += 128) {
        int cls = i / OCC, o = i % OCC;
        sLW[i] = lw[(size_t)cls * FLAT + o * NLOC + hw];
    }
    for (int i = tid; i < OCC; i += 128)
        sB[i] = bias[i * NLOC + hw];

    // Third N-tile covers N=32..47; columns >= 42 of D are discarded, so the
    // B fragment there may hold any in-bounds value. Clamp once.
    const int n2  = mrow + 32;
    const int n2c = (n2 < OCC) ? n2 : (OCC - 1);

    // --- batch chunks ------------------------------------------------------
    for (int chunk = 0; chunk < NCHUNK; ++chunk) {
        __syncthreads();  // sAct/sP free to be overwritten; sW ready (1st iter)

        // gather patch matrix: sP[bi][kg] = x[b][c][h+ki][w+kj]
        for (int i = tid; i < CHUNK * KPAD; i += 128) {
            int bi = i / KPAD, kg = i % KPAD;
            float v = 0.0f;
            if (kg < KDIM) {
                int c = kg / 49, r = kg % 49;
                int ki = r / 7, kj = r % 7;
                int b = chunk * CHUNK + bi;
                v = x[(((size_t)b * ICC + c) * 32 + (h + ki)) * 32 + (w + kj)];
            }
            sP[i] = v;
        }
        __syncthreads();

        // WMMA: out[64,48] = P[64,148] x W^T[148,48]; each wave one 16-row M-tile
        v8f acc0 = {}, acc1 = {}, acc2 = {};
        const float* prow = sP + (wv * 16 + mrow) * KPAD;
        for (int ks = 0; ks < KPAD; ks += 4) {
            int k = ks + khalf * 2;                    // even -> 8B aligned
            v2f a = *(const v2f*)(prow + k);           // A[M=mrow][k..k+1]
            const float* wp = sW + (k >> 1) * (OCC * 2);
            v2f b0 = *(const v2f*)(wp + 2 * mrow);
            v2f b1 = *(const v2f*)(wp + 2 * (mrow + 16));
            v2f b2 = *(const v2f*)(wp + 2 * n2c);
            acc0 = __builtin_amdgcn_wmma_f32_16x16x4_f32(
                       false, a, false, b0, (short)0, acc0, false, false);
            acc1 = __builtin_amdgcn_wmma_f32_16x16x4_f32(
                       false, a, false, b1, (short)0, acc1, false, false);
            acc2 = __builtin_amdgcn_wmma_f32_16x16x4_f32(
                       false, a, false, b2, (short)0, acc2, false, false);
        }
        __syncthreads();  // everyone done reading sP; reuse as sAct

        // bias + ReLU, scatter accumulators into sAct[row][o]
        // C/D layout: VGPR r -> M = khalf*8 + r, N = mrow + 16*tile
        #pragma unroll
        for (int r = 0; r < 8; ++r) {
            int row = wv * 16 + khalf * 8 + r;
            float v0 = acc0[r] + sB[mrow];
            sAct[row * OCC + mrow] = v0 > 0.0f ? v0 : 0.0f;
            float v1 = acc1[r] + sB[mrow + 16];
            sAct[row * OCC + mrow + 16] = v1 > 0.0f ? v1 : 0.0f;
            if (n2 < OCC) {
                float v2 = acc2[r] + sB[n2];
                sAct[row * OCC + n2] = v2 > 0.0f ? v2 : 0.0f;
            }
        }
        __syncthreads();

        // per-location classifier partials: 64 rows x 10 classes
        for (int i = tid; i < CHUNK * NCLS; i += 128) {
            int bi = i / NCLS, cls = i % NCLS;
            const float* ar = sAct + bi * OCC;
            const float* lr = sLW + cls * OCC;
            float s = 0.0f;
            #pragma unroll
            for (int o = 0; o < OCC; ++o) s += ar[o] * lr[o];
            partial[((size_t)hw * NB + chunk * CHUNK + bi) * NCLS + cls] = s;
        }
    }
}

// Deterministic reduction over the 676 location slabs + lb
__global__ void lc_reduce_kernel(const float* __restrict__ partial,
                                 const float* __restrict__ lb,
                                 float* __restrict__ out)
{
    int i = blockIdx.x * blockDim.x + threadIdx.x;   // 0..10239
    if (i >= NB * NCLS) return;
    float s = lb[i % NCLS];
    for (int hw = 0; hw < NLOC; ++hw)
        s += partial[(size_t)hw * NB * NCLS + i];
    out[i] = s;
}

extern "C" void kernel_launch(void* const* d_in, const int* in_sizes, int n_in,
                              void* d_out, int out_size, void* d_ws, size_t ws_size,
                              hipStream_t stream) {
    (void)in_sizes; (void)n_in; (void)out_size; (void)ws_size;
    const float* x    = (const float*)d_in[0];
    const float* W    = (const float*)d_in[1];
    const float* bias = (const float*)d_in[2];
    const float* lw   = (const float*)d_in[3];
    const float* lb   = (const float*)d_in[4];
    float* out     = (float*)d_out;
    float* partial = (float*)d_ws;   // 676*1024*10 floats = 27.7 MB

    lc_wmma_kernel<<<NLOC, 128, 0, stream>>>(x, W, bias, lw, partial);
    lc_reduce_kernel<<<(NB * NCLS + 255) / 256, 256, 0, stream>>>(partial, lb, out);
}